// UnweightedHistogramSingularityStrengthRecalibration_85409719648430
// MI455X (gfx1250) — compile-verified
//
#include <hip/hip_runtime.h>
#include <math.h>

// ---- problem constants (from reference setup_inputs) ----
#define Hdim 256
#define Wdim 256
#define Cdim 32
#define Kdim 8

// ---- tiling ----
#define TILE_H 32
#define TILE_W 8
#define LH (TILE_H + 6)   // 38 rows incl. radius-3 halo
#define LW (TILE_W + 6)   // 14 cols incl. radius-3 halo

// least-squares slope weights: (log(2r+1) - mean) / sum((log-mean)^2), r=1..3
#define W1f (-1.2436233f)
#define W2f ( 0.1596544f)
#define W3f ( 1.0839688f)

typedef int v4i __attribute__((ext_vector_type(4)));
typedef __attribute__((address_space(1))) v4i* g_v4i_ptr;   // global (HIP prints as __device__)
typedef __attribute__((address_space(3))) v4i* l_v4i_ptr;   // LDS

#if __has_builtin(__builtin_amdgcn_global_load_async_to_lds_b128)
#define HAVE_ASYNC_LDS 1
template <int OFF>
__device__ __forceinline__ void async_g2l_b128(const float* gsrc, float* ldst) {
  // INST_OFFSET (immediate) is added to BOTH the global and LDS address (ISA 08 §4.4)
  __builtin_amdgcn_global_load_async_to_lds_b128(
      (g_v4i_ptr)const_cast<float*>(gsrc), (l_v4i_ptr)ldst, OFF, 0);
}
__device__ __forceinline__ void async_copy_pixel128B(const float* gsrc, float* ldst) {
  async_g2l_b128<  0>(gsrc, ldst);
  async_g2l_b128< 16>(gsrc, ldst);
  async_g2l_b128< 32>(gsrc, ldst);
  async_g2l_b128< 48>(gsrc, ldst);
  async_g2l_b128< 64>(gsrc, ldst);
  async_g2l_b128< 80>(gsrc, ldst);
  async_g2l_b128< 96>(gsrc, ldst);
  async_g2l_b128<112>(gsrc, ldst);
}
#else
#warning "gfx1250 probe: __builtin_amdgcn_global_load_async_to_lds_b128 NOT available; using sync fallback"
#define HAVE_ASYNC_LDS 0
#endif

__device__ __forceinline__ void wait_async_zero() {
#if __has_builtin(__builtin_amdgcn_s_wait_asynccnt)
  __builtin_amdgcn_s_wait_asynccnt(0);
#else
  asm volatile("s_wait_asynccnt 0" ::: "memory");
#endif
}

__global__ __launch_bounds__(256)
void UnweightedHistogramSingularityStrengthRecalibration_kernel(
    const float* __restrict__ x, const float* __restrict__ centers,
    const float* __restrict__ widths, float* __restrict__ out) {
  __shared__ __align__(16) float tile[LH * LW * Cdim];  // 68,096 B (<< 320KB/WGP)

  const int tid  = threadIdx.x;
  const int lane = tid & 31;   // channel index (wave32: lane == channel)
  const int wv   = tid >> 5;   // wave id == output column within tile
  const int gw0  = blockIdx.x * TILE_W;
  const int gh0  = blockIdx.y * TILE_H;
  const int b    = blockIdx.z;

  // ---------------- stage halo tile into LDS (async DMA path) ----------------
  for (int p = tid; p < LH * LW; p += 256) {
    const int lrow = p / LW;
    const int lcol = p - lrow * LW;
    const int gh = gh0 - 3 + lrow;
    const int gw = gw0 - 3 + lcol;
    float* ldst = &tile[p * Cdim];
    if (((unsigned)gh < (unsigned)Hdim) && ((unsigned)gw < (unsigned)Wdim)) {
      const float* gsrc = x + (size_t)(((b * Hdim + gh) * Wdim + gw)) * Cdim;
#if HAVE_ASYNC_LDS
      async_copy_pixel128B(gsrc, ldst);
#else
      #pragma unroll
      for (int q = 0; q < 8; ++q) ((float4*)ldst)[q] = ((const float4*)gsrc)[q];
#endif
    } else {
      const float4 z = make_float4(0.f, 0.f, 0.f, 0.f);  // SAME padding uses 0
      #pragma unroll
      for (int q = 0; q < 8; ++q) ((float4*)ldst)[q] = z;
    }
  }

  // ---- per-channel soft-histogram params: overlap with the async DMA -------
  float cen[Kdim], wid[Kdim];
  {
    const float4* c4 = (const float4*)(centers + lane * Kdim);
    const float4* w4 = (const float4*)(widths  + lane * Kdim);
    float4 a = c4[0], bq = c4[1], u = w4[0], v = w4[1];
    cen[0]=a.x; cen[1]=a.y; cen[2]=a.z; cen[3]=a.w;
    cen[4]=bq.x; cen[5]=bq.y; cen[6]=bq.z; cen[7]=bq.w;
    wid[0]=u.x; wid[1]=u.y; wid[2]=u.z; wid[3]=u.w;
    wid[4]=v.x; wid[5]=v.y; wid[6]=v.z; wid[7]=v.w;
  }

#if HAVE_ASYNC_LDS
  wait_async_zero();
#endif
  __syncthreads();

  // LDS accessor: row r (0..LH-1), col cc (0..13), this lane's channel
  #define TLD(r, cc) tile[(((r) * LW) + (cc)) * Cdim + lane]

  // ---------------- rolling separable box sums, wave sweeps one column -------
  // During iteration oh: h7[i] = hsum7(row oh+i), same for h5/h3.
  float h3[7], h5[7], h7[7];
  #pragma unroll
  for (int j = 0; j < 6; ++j) {   // prime rows 0..5 into slots 1..6
    float x0=TLD(j,wv+0), x1=TLD(j,wv+1), x2=TLD(j,wv+2), x3=TLD(j,wv+3),
          x4=TLD(j,wv+4), x5=TLD(j,wv+5), x6=TLD(j,wv+6);
    float s3 = x2 + x3 + x4;
    float s5 = s3 + x1 + x5;
    float s7 = s5 + x0 + x6;
    h3[j+1]=s3; h5[j+1]=s5; h7[j+1]=s7;
  }

  #pragma unroll 4
  for (int oh = 0; oh < TILE_H; ++oh) {
    #pragma unroll
    for (int i = 0; i < 6; ++i) { h3[i]=h3[i+1]; h5[i]=h5[i+1]; h7[i]=h7[i+1]; }
    const int r = oh + 6;
    {
      float x0=TLD(r,wv+0), x1=TLD(r,wv+1), x2=TLD(r,wv+2), x3=TLD(r,wv+3),
            x4=TLD(r,wv+4), x5=TLD(r,wv+5), x6=TLD(r,wv+6);
      float s3 = x2 + x3 + x4;
      float s5 = s3 + x1 + x5;
      float s7 = s5 + x0 + x6;
      h3[6]=s3; h5[6]=s5; h7[6]=s7;
    }
    const float mu3 = h3[2] + h3[3] + h3[4];
    const float mu5 = h5[1] + h5[2] + h5[3] + h5[4] + h5[5];
    const float mu7 = h7[0] + h7[1] + h7[2] + h7[3] + h7[4] + h7[5] + h7[6];

    const float lm1 = __logf(mu3 + 1e-6f);
    const float lm2 = __logf(mu5 + 1e-6f);
    const float lm3 = __logf(mu7 + 1e-6f);
    const float alpha = fmaf(W1f, lm1, fmaf(W2f, lm2, W3f * lm3));

    float sc = 0.0f;
    #pragma unroll
    for (int k = 0; k < Kdim; ++k) {
      const float d = fabsf(alpha - cen[k]);
      const float m = fmaf(-d, wid[k], 1.0f);   // 1 - |alpha-c|*w
      sc += fmaxf(m, 0.0f);
    }
    const float sig = __fdividef(1.0f, 1.0f + __expf(-sc));

    const float xc = TLD(oh + 3, wv + 3);
    // output is write-once / never re-read: non-temporal store (TH=NT)
    __builtin_nontemporal_store(
        xc + sig,
        &out[(size_t)(((b * Hdim + (gh0 + oh)) * Wdim) + (gw0 + wv)) * Cdim + lane]);
  }
  #undef TLD
}

extern "C" void kernel_launch(void* const* d_in, const int* in_sizes, int n_in,
                              void* d_out, int out_size, void* d_ws, size_t ws_size,
                              hipStream_t stream) {
  const float* x       = (const float*)d_in[0];
  const float* centers = (const float*)d_in[1];
  const float* widths  = (const float*)d_in[2];
  float* out = (float*)d_out;

  const int B = in_sizes[0] / (Hdim * Wdim * Cdim);  // 8
  dim3 grid(Wdim / TILE_W, Hdim / TILE_H, B);        // 32 x 8 x 8 = 2048 blocks
  dim3 block(256);
  hipLaunchKernelGGL(UnweightedHistogramSingularityStrengthRecalibration_kernel,
                     grid, block, 0, stream, x, centers, widths, out);
}